// GraphEncoder_52836687675443
// MI455X (gfx1250) — compile-verified
//
#include <hip/hip_runtime.h>
#include <hip/hip_bf16.h>
#include <math.h>

// ---------------- problem constants (match reference) ----------------
#define NN 100000      // nodes
#define NE 600000      // edges
#define HH 64          // hidden
#define NGRAPH 512

typedef __attribute__((ext_vector_type(2))) float v2f;
typedef __attribute__((ext_vector_type(8))) float v8f;
typedef __attribute__((ext_vector_type(4))) unsigned int v4u;
typedef __attribute__((ext_vector_type(4))) int v4i;
typedef __attribute__((ext_vector_type(8))) int v8i;

// ---------------- CDNA5 WMMA fp32 16x16x4 ----------------
// A (16x4 f32): lanes 0-15 hold row M=lane, K=k0..k0+1 ; lanes 16-31 hold K=k0+2..k0+3
// B (4x16 f32): lanes 0-15 hold col N=lane for K=k0..k0+1 ; lanes 16-31 K=k0+2..3
// C/D (16x16 f32): VGPR i: lanes 0-15 -> M=i, lanes 16-31 -> M=i+8, N=lane&15
__device__ __forceinline__ v8f wmma4(v2f a, v2f b, v8f c) {
  return __builtin_amdgcn_wmma_f32_16x16x4_f32(
      /*neg_a=*/false, a, /*neg_b=*/false, b,
      /*c_mod=*/(short)0, c, /*reuse_a=*/false, /*reuse_b=*/false);
}

// ---------------- TDM: DMA a (K x 64) f32 panel of B into LDS ----------------
// Builds Tensor DMA Descriptor per CDNA5 ISA:
//  group0: [1:0]=count=1, [63:32]=lds_addr, [120:64]=global_addr, [127:126]=type=2
//  group1: [17:16]=data_size(2 -> 4B), [79:48]=tensor_dim0, [111:80]=tensor_dim1,
//          [127:112]=tile_dim0, [143:128]=tile_dim1, [207:160]=tensor_dim0_stride
// Issued by one wave; completion via TENSORcnt.
__device__ __forceinline__ void tdm_load_panel(float* lds_dst, const float* gsrc,
                                               unsigned ldb, unsigned K) {
  unsigned lds = (unsigned)(size_t)(void*)lds_dst;   // low 32 bits = LDS byte offset
  unsigned long long ga = (unsigned long long)(size_t)(const void*)gsrc;
  v4u g0;
  g0[0] = 1u;                                        // count=1, user descriptor
  g0[1] = lds;                                       // lds_addr
  g0[2] = (unsigned)ga;                              // global_addr[31:0]
  g0[3] = (unsigned)((ga >> 32) & 0x01FFFFFFull) | (2u << 30);   // addr[56:32] | type=2
  unsigned d0 = ldb;                                 // tensor_dim0 (row length, elems)
  unsigned d1 = K;                                   // tensor_dim1 (#rows)
  v8i g1;
  g1[0] = (int)(2u << 16);                           // data_size = 4 bytes
  g1[1] = (int)((d0 & 0xFFFFu) << 16);               // [47:32]=0, tensor_dim0 lo16
  g1[2] = (int)((d0 >> 16) | ((d1 & 0xFFFFu) << 16));// dim0 hi16 | dim1 lo16
  g1[3] = (int)((d1 >> 16) | (64u << 16));           // dim1 hi16 | tile_dim0=64
  g1[4] = (int)(K & 0xFFFFu);                        // tile_dim1=K, tile_dim2=0
  g1[5] = (int)ldb;                                  // tensor_dim0_stride lo32
  g1[6] = 0;                                         // stride hi16 | dim1_stride lo16
  g1[7] = 0;
  v4i gz = {0, 0, 0, 0};
#if __clang_major__ >= 23
  v8i gz8 = {0, 0, 0, 0, 0, 0, 0, 0};
  __builtin_amdgcn_tensor_load_to_lds(g0, g1, gz, gz, gz8, 0);
#else
  __builtin_amdgcn_tensor_load_to_lds(g0, g1, gz, gz, 0);
#endif
  __builtin_amdgcn_s_wait_tensorcnt((short)0);
}

// monotone float<->uint encoding for atomicMax on floats
__device__ __forceinline__ unsigned encf(float f) {
  unsigned u = __float_as_uint(f);
  return (u & 0x80000000u) ? ~u : (u | 0x80000000u);
}
__device__ __forceinline__ float decf(unsigned u) {
  u = (u & 0x80000000u) ? (u & 0x7FFFFFFFu) : ~u;
  return __uint_as_float(u);
}
__device__ __forceinline__ float sigmoidf_(float x) { return 1.0f / (1.0f + __expf(-x)); }

__device__ __forceinline__ int lower_bound_dev(const int* a, int n, int key) {
  int lo = 0, hi = n;
  while (lo < hi) { int mid = (lo + hi) >> 1; if (a[mid] < key) lo = mid + 1; else hi = mid; }
  return lo;
}

// =====================================================================
// K0: init attention accumulators
// =====================================================================
__global__ void init_att(unsigned* nmax, float* nsum, float* oat) {
  size_t t = (size_t)blockIdx.x * 256 + threadIdx.x;
  if (t < (size_t)NN * 4) { nmax[t] = encf(-3.0e38f); nsum[t] = 0.0f; }
  if (t < (size_t)NN * 256) oat[t] = 0.0f;
}

// =====================================================================
// K1: hx = LN(x_s @ node_W + node_b)   (WMMA + LDS-tile LayerNorm)
// block = 64 threads (2 waves), each wave -> 16 nodes
// =====================================================================
__global__ void node_pre(const float* __restrict__ x, const float* __restrict__ W,
                         const float* __restrict__ bias,
                         const float* __restrict__ lg, const float* __restrict__ lb,
                         float* __restrict__ hx) {
  __shared__ float tile[2][16][64];
  int lane = threadIdx.x & 31, wave = threadIdx.x >> 5;
  int m0 = (blockIdx.x * 2 + wave) * 16;
  int r = lane & 15, hv = lane >> 4;
  v8f acc[4] = {};
  for (int k0 = 0; k0 < 32; k0 += 4) {
    v2f a; const float* ap = x + (size_t)(m0 + r) * 32 + k0 + 2 * hv;
    a.x = ap[0]; a.y = ap[1];
    for (int t = 0; t < 4; ++t) {
      v2f b; const float* bp = W + (size_t)(k0 + 2 * hv) * 64 + t * 16 + r;
      b.x = bp[0]; b.y = bp[64];
      acc[t] = wmma4(a, b, acc[t]);
    }
  }
  for (int t = 0; t < 4; ++t)
    for (int i = 0; i < 8; ++i)
      tile[wave][i + 8 * hv][t * 16 + r] = acc[t][i] + bias[t * 16 + r];
  __syncthreads();
  if (lane < 16) {
    float* row = tile[wave][lane];
    float m = 0.f; for (int c = 0; c < 64; ++c) m += row[c]; m *= (1.0f / 64.0f);
    float va = 0.f; for (int c = 0; c < 64; ++c) { float d = row[c] - m; va += d * d; }
    va *= (1.0f / 64.0f);
    float rs = rsqrtf(va + 1e-5f);
    float* o = hx + (size_t)(m0 + lane) * 64;
    for (int c = 0; c < 64; ++c) o[c] = (row[c] - m) * rs * lg[c] + lb[c];
  }
}

// =====================================================================
// K2: generic D = A(M x K) @ B(K x Ncols) + bias  via WMMA
// B panel (K x 64) staged into LDS by the Tensor Data Mover (one DMA per block)
// grid = (M/16, Ncols/64), block = 128 (4 waves, one 16-wide n-tile each), K<=64
// =====================================================================
__global__ void gemm_bias(const float* __restrict__ A, const float* __restrict__ B,
                          const float* __restrict__ bias, float* __restrict__ D,
                          int lda, int ldb, int K) {
  __shared__ float Bs[64 * 64];                 // K x 64 panel, row stride 64
  int lane = threadIdx.x & 31, wave = threadIdx.x >> 5;
  int m0 = blockIdx.x * 16;
  int nblk = blockIdx.y * 64;
  if (wave == 0) {                              // one TDM op per workgroup
    tdm_load_panel(Bs, B + nblk, (unsigned)ldb, (unsigned)K);
  }
  __syncthreads();
  int r = lane & 15, hv = lane >> 4;
  int n0l = wave * 16;
  v8f acc = {};
  for (int k0 = 0; k0 < K; k0 += 4) {
    v2f a; const float* ap = A + (size_t)(m0 + r) * lda + k0 + 2 * hv;
    a.x = ap[0]; a.y = ap[1];
    v2f b; const float* bp = Bs + (k0 + 2 * hv) * 64 + n0l + r;
    b.x = bp[0]; b.y = bp[64];
    acc = wmma4(a, b, acc);
  }
  for (int i = 0; i < 8; ++i) {
    int row = i + 8 * hv;
    D[(size_t)(m0 + row) * ldb + nblk + n0l + r] = acc[i] + bias[nblk + n0l + r];
  }
}

// =====================================================================
// K3: edge pipeline: feat = emb + tiled posenc ; he = LN(feat@edge_W+b) ;
//     ee = he @ We + be.  block = 64 (2 waves, 16 edges each)
// =====================================================================
#define LOG1EM4 (-9.210340371976184f)
__global__ void edge_feat(const int* __restrict__ ea, const float* __restrict__ emb,
                          const float* __restrict__ eW, const float* __restrict__ eb,
                          const float* __restrict__ lg, const float* __restrict__ lb,
                          const float* __restrict__ We, const float* __restrict__ be2,
                          float* __restrict__ ee) {
  __shared__ float feat[2][16][192];
  __shared__ float het[2][16][64];
  int lane = threadIdx.x & 31, wave = threadIdx.x >> 5;
  int e0 = (blockIdx.x * 2 + wave) * 16;
  // build 16x192 input tile: embedding gather + positional encoding (tiled 3x)
  for (int t = lane; t < 16 * 192; t += 32) {
    int j = t / 192, c = t - j * 192;
    int tb = c >> 6, ph = c & 63;
    const int* ap = ea + (size_t)(e0 + j) * 4;
    float pos = (float)ap[0];
    float fr = __expf(LOG1EM4 * (float)(ph & 31) * (1.0f / 31.0f));
    float pe = (ph < 32) ? __sinf(pos * fr) : __cosf(pos * fr);
    int idx = ap[1 + tb];
    float ev = emb[((size_t)tb * 10000 + idx) * 64 + ph];
    feat[wave][j][c] = ev + pe;
  }
  __syncthreads();
  int r = lane & 15, hv = lane >> 4;
  {
    v8f acc[4] = {};
    for (int k0 = 0; k0 < 192; k0 += 4) {
      v2f a; a.x = feat[wave][r][k0 + 2 * hv]; a.y = feat[wave][r][k0 + 2 * hv + 1];
      for (int t = 0; t < 4; ++t) {
        v2f b; const float* bp = eW + (size_t)(k0 + 2 * hv) * 64 + t * 16 + r;
        b.x = bp[0]; b.y = bp[64];
        acc[t] = wmma4(a, b, acc[t]);
      }
    }
    for (int t = 0; t < 4; ++t)
      for (int i = 0; i < 8; ++i)
        het[wave][i + 8 * hv][t * 16 + r] = acc[t][i] + eb[t * 16 + r];
  }
  __syncthreads();
  if (lane < 16) {   // LayerNorm rows of he in LDS
    float* row = het[wave][lane];
    float m = 0.f; for (int c = 0; c < 64; ++c) m += row[c]; m *= (1.0f / 64.0f);
    float va = 0.f; for (int c = 0; c < 64; ++c) { float d = row[c] - m; va += d * d; }
    va *= (1.0f / 64.0f);
    float rs = rsqrtf(va + 1e-5f);
    for (int c = 0; c < 64; ++c) row[c] = (row[c] - m) * rs * lg[c] + lb[c];
  }
  __syncthreads();
  for (int nb = 0; nb < 4; ++nb) {   // ee = he @ We (64 -> 256)
    v8f acc[4] = {};
    for (int k0 = 0; k0 < 64; k0 += 4) {
      v2f a; a.x = het[wave][r][k0 + 2 * hv]; a.y = het[wave][r][k0 + 2 * hv + 1];
      for (int t = 0; t < 4; ++t) {
        int n0 = nb * 64 + t * 16;
        v2f b; const float* bp = We + (size_t)(k0 + 2 * hv) * 256 + n0 + r;
        b.x = bp[0]; b.y = bp[256];
        acc[t] = wmma4(a, b, acc[t]);
      }
    }
    for (int t = 0; t < 4; ++t)
      for (int i = 0; i < 8; ++i) {
        int n0 = nb * 64 + t * 16;
        ee[(size_t)(e0 + i + 8 * hv) * 256 + n0 + r] = acc[t][i] + be2[n0 + r];
      }
  }
}

// =====================================================================
// K4a/b/c: segment softmax attention + message scatter
// =====================================================================
__global__ void att_raw(const float* __restrict__ q, const float* __restrict__ k,
                        const float* __restrict__ ee, const int* __restrict__ ei,
                        float* __restrict__ alp, unsigned* __restrict__ nmax) {
  int t = blockIdx.x * 256 + threadIdx.x;      // t < NE*4 (exact)
  int e = t >> 2, hv = t & 3;
  int src = ei[e], dst = ei[NE + e];
  const float* qp = q + (size_t)dst * 256 + hv * 64;
  const float* kp = k + (size_t)src * 256 + hv * 64;
  const float* ep = ee + (size_t)e * 256 + hv * 64;
  __builtin_prefetch(qp, 0, 0);                // global_prefetch_b8
  __builtin_prefetch(kp, 0, 0);
  float s = 0.f;
  for (int c = 0; c < 64; ++c) s += qp[c] * (kp[c] + ep[c]);
  s *= 0.125f;                                  // 1/sqrt(64)
  alp[t] = s;
  atomicMax(nmax + (size_t)dst * 4 + hv, encf(s));
}

__global__ void att_exp(const int* __restrict__ ei, const unsigned* __restrict__ nmax,
                        float* __restrict__ alp, float* __restrict__ nsum) {
  int t = blockIdx.x * 256 + threadIdx.x;
  int e = t >> 2, hv = t & 3;
  int dst = ei[NE + e];
  float mx = decf(nmax[(size_t)dst * 4 + hv]);
  float ex = __expf(alp[t] - mx);
  alp[t] = ex;
  atomicAdd(nsum + (size_t)dst * 4 + hv, ex);
}

__global__ void att_scatter(const float* __restrict__ vv, const float* __restrict__ ee,
                            const float* __restrict__ alp, const float* __restrict__ nsum,
                            const int* __restrict__ ei, float* __restrict__ oat) {
  int e = blockIdx.x;
  int c = threadIdx.x;
  int hv = c >> 6;
  int src = ei[e], dst = ei[NE + e];
  float coef = alp[(size_t)e * 4 + hv] / (nsum[(size_t)dst * 4 + hv] + 1e-16f);
  float val = (vv[(size_t)src * 256 + c] + ee[(size_t)e * 256 + c]) * coef;
  atomicAdd(oat + (size_t)dst * 256 + c, val);
}

// =====================================================================
// K5a: beta gate. block=256 (8 waves), one wave per node (wave32 shuffle reduce)
// =====================================================================
__global__ void beta_gate(const float* __restrict__ oat, const float* __restrict__ xr,
                          const float* __restrict__ wb, float* __restrict__ gbuf) {
  int lane = threadIdx.x & 31, wave = threadIdx.x >> 5;
  size_t nd = (size_t)blockIdx.x * 8 + wave;
  const float* op = oat + nd * 256;
  const float* xp = xr + nd * 256;
  float part = 0.f;
  for (int i = 0; i < 8; ++i) {
    int c = lane + 32 * i;
    float o = op[c], x2 = xp[c];
    part += o * wb[c] + x2 * wb[256 + c] + (o - x2) * wb[512 + c];
  }
  for (int off = 16; off; off >>= 1) part += __shfl_xor(part, off, 32);
  float beta = sigmoidf_(part);
  float* gp = gbuf + nd * 256;
  for (int i = 0; i < 8; ++i) {
    int c = lane + 32 * i;
    gp[c] = beta * xp[c] + (1.0f - beta) * op[c];
  }
}

// =====================================================================
// K5b: h = GELU(LN(g@lin1_W+b)) + hx ; zin = LN(h)  (fused in LDS)
// block = 64 (2 waves)
// =====================================================================
__global__ void node_post(const float* __restrict__ gbuf, const float* __restrict__ W1,
                          const float* __restrict__ b1, const float* __restrict__ g1,
                          const float* __restrict__ bb1, const float* __restrict__ ng,
                          const float* __restrict__ nb2, const float* __restrict__ hx,
                          float* __restrict__ zin) {
  __shared__ float tile[2][16][64];
  int lane = threadIdx.x & 31, wave = threadIdx.x >> 5;
  int m0 = (blockIdx.x * 2 + wave) * 16;
  int r = lane & 15, hv = lane >> 4;
  v8f acc[4] = {};
  for (int k0 = 0; k0 < 256; k0 += 4) {
    v2f a; const float* ap = gbuf + (size_t)(m0 + r) * 256 + k0 + 2 * hv;
    a.x = ap[0]; a.y = ap[1];
    for (int t = 0; t < 4; ++t) {
      v2f b; const float* bp = W1 + (size_t)(k0 + 2 * hv) * 64 + t * 16 + r;
      b.x = bp[0]; b.y = bp[64];
      acc[t] = wmma4(a, b, acc[t]);
    }
  }
  for (int t = 0; t < 4; ++t)
    for (int i = 0; i < 8; ++i)
      tile[wave][i + 8 * hv][t * 16 + r] = acc[t][i] + b1[t * 16 + r];
  __syncthreads();
  if (lane < 16) {
    float* row = tile[wave][lane];
    float m = 0.f; for (int c = 0; c < 64; ++c) m += row[c]; m *= (1.0f / 64.0f);
    float va = 0.f; for (int c = 0; c < 64; ++c) { float d = row[c] - m; va += d * d; }
    va *= (1.0f / 64.0f);
    float rs = rsqrtf(va + 1e-5f);
    const float* hp = hx + (size_t)(m0 + lane) * 64;
    for (int c = 0; c < 64; ++c) {
      float t2 = (row[c] - m) * rs * g1[c] + bb1[c];
      t2 = 0.5f * t2 * (1.0f + erff(t2 * 0.70710678118f));   // exact GELU
      row[c] = t2 + hp[c];
    }
    float m2 = 0.f; for (int c = 0; c < 64; ++c) m2 += row[c]; m2 *= (1.0f / 64.0f);
    float v2 = 0.f; for (int c = 0; c < 64; ++c) { float d = row[c] - m2; v2 += d * d; }
    v2 *= (1.0f / 64.0f);
    float rs2 = rsqrtf(v2 + 1e-5f);
    float* zo = zin + (size_t)(m0 + lane) * 64;
    for (int c = 0; c < 64; ++c) zo[c] = (row[c] - m2) * rs2 * ng[c] + nb2[c];
  }
}

// =====================================================================
// K6: Set2Set — one block per graph (graphs independent; batch_s sorted)
// block = 128
// =====================================================================
__global__ void set2set_k(const float* __restrict__ z, const int* __restrict__ batch,
                          const float* __restrict__ Wih, const float* __restrict__ Whh,
                          const float* __restrict__ bih, const float* __restrict__ bhh,
                          float* __restrict__ outp) {
  __shared__ float hS[64], cS[64], qstar[128], gates[256], rS[64], red[128];
  __shared__ float sS;
  __shared__ int sStart, sEnd;
  int tid = threadIdx.x;
  int g = blockIdx.x;
  if (tid == 0) {
    sStart = lower_bound_dev(batch, NN, g);
    sEnd   = lower_bound_dev(batch, NN, g + 1);
  }
  if (tid < 64) { hS[tid] = 0.f; cS[tid] = 0.f; }
  qstar[tid] = 0.f;
  __syncthreads();
  int start = sStart, end = sEnd;

  for (int step = 0; step < 4; ++step) {
    // LSTM gates: i,f,g,o
    for (int j = tid; j < 256; j += 128) {
      float acc = bih[j] + bhh[j];
      for (int k2 = 0; k2 < 128; ++k2) acc += qstar[k2] * Wih[(size_t)j * 128 + k2];
      for (int k2 = 0; k2 < 64;  ++k2) acc += hS[k2]   * Whh[(size_t)j * 64 + k2];
      gates[j] = acc;
    }
    __syncthreads();
    if (tid < 64) {
      float ig = sigmoidf_(gates[tid]);
      float fg = sigmoidf_(gates[64 + tid]);
      float gg = tanhf(gates[128 + tid]);
      float og = sigmoidf_(gates[192 + tid]);
      float cn = fg * cS[tid] + ig * gg;
      cS[tid] = cn;
      hS[tid] = og * tanhf(cn);
    }
    __syncthreads();
    // segment softmax over this graph's nodes: max pass
    float lmax = -3.0e38f;
    for (int nd = start + tid; nd < end; nd += 128) {
      const float* zp = z + (size_t)nd * 64;
      float en = 0.f; for (int k2 = 0; k2 < 64; ++k2) en += zp[k2] * hS[k2];
      lmax = fmaxf(lmax, en);
    }
    red[tid] = lmax;
    __syncthreads();
    for (int off = 64; off > 0; off >>= 1) {
      if (tid < off) red[tid] = fmaxf(red[tid], red[tid + off]);
      __syncthreads();
    }
    float mx = red[0];
    if (tid < 64) rS[tid] = 0.f;
    if (tid == 0) sS = 0.f;
    __syncthreads();
    // exp + weighted sum
    for (int nd = start + tid; nd < end; nd += 128) {
      const float* zp = z + (size_t)nd * 64;
      float en = 0.f; for (int k2 = 0; k2 < 64; ++k2) en += zp[k2] * hS[k2];
      float ex = __expf(en - mx);
      atomicAdd(&sS, ex);
      for (int k2 = 0; k2 < 64; ++k2) atomicAdd(&rS[k2], ex * zp[k2]);
    }
    __syncthreads();
    if (tid < 64) {
      qstar[tid] = hS[tid];
      qstar[64 + tid] = rS[tid] / (sS + 1e-16f);
    }
    __syncthreads();
  }
  outp[(size_t)g * 128 + tid] = qstar[tid];
}

// =====================================================================
// host launcher
// =====================================================================
extern "C" void kernel_launch(void* const* d_in, const int* in_sizes, int n_in,
                              void* d_out, int out_size, void* d_ws, size_t ws_size,
                              hipStream_t stream) {
  (void)in_sizes; (void)n_in; (void)out_size; (void)ws_size;
  // inputs
  const float* x_s   = (const float*)d_in[0];    // (N,32)
  const int*   ea    = (const int*)d_in[1];      // (E,4)
  const int*   ei    = (const int*)d_in[2];      // (2,E)
  const int*   batch = (const int*)d_in[3];      // (N,)
  // params (dict insertion order)
  const float* node_W   = (const float*)d_in[4];
  const float* node_b   = (const float*)d_in[5];
  const float* node_lng = (const float*)d_in[6];
  const float* node_lnb = (const float*)d_in[7];
  const float* emb      = (const float*)d_in[8];   // (3,10000,64)
  const float* edge_W   = (const float*)d_in[9];   // (192,64)
  const float* edge_b   = (const float*)d_in[10];
  const float* edge_lng = (const float*)d_in[11];
  const float* edge_lnb = (const float*)d_in[12];
  const float* Wq = (const float*)d_in[13]; const float* bq = (const float*)d_in[14];
  const float* Wk = (const float*)d_in[15]; const float* bk = (const float*)d_in[16];
  const float* Wv = (const float*)d_in[17]; const float* bv = (const float*)d_in[18];
  const float* We = (const float*)d_in[19]; const float* be = (const float*)d_in[20];
  const float* Wskip = (const float*)d_in[21]; const float* bskip = (const float*)d_in[22];
  const float* Wbeta = (const float*)d_in[23];     // (768,1)
  const float* lin1_W = (const float*)d_in[24];    // (256,64)
  const float* lin1_b = (const float*)d_in[25];
  const float* ln1_g  = (const float*)d_in[26];
  const float* ln1_b  = (const float*)d_in[27];
  const float* norm_g = (const float*)d_in[28];
  const float* norm_b = (const float*)d_in[29];
  const float* lin_W  = (const float*)d_in[30];    // (64,64)
  const float* lin_b  = (const float*)d_in[31];
  const float* lstm_Wih = (const float*)d_in[32];  // (256,128)
  const float* lstm_Whh = (const float*)d_in[33];  // (256,64)
  const float* lstm_bih = (const float*)d_in[34];
  const float* lstm_bhh = (const float*)d_in[35];

  // workspace carve
  float* hx  = (float*)d_ws;                       // N*64
  float* q   = hx  + (size_t)NN * 64;              // N*256 (later reused as g-buffer)
  float* kb  = q   + (size_t)NN * 256;             // N*256
  float* vb  = kb  + (size_t)NN * 256;             // N*256
  float* xr  = vb  + (size_t)NN * 256;             // N*256
  float* ee  = xr  + (size_t)NN * 256;             // E*256
  float* alp = ee  + (size_t)NE * 256;             // E*4
  unsigned* nmax = (unsigned*)(alp + (size_t)NE * 4); // N*4
  float* nsum = (float*)(nmax + (size_t)NN * 4);   // N*4
  float* oat  = nsum + (size_t)NN * 4;             // N*256
  float* zin  = oat  + (size_t)NN * 256;           // N*64
  float* zz   = zin  + (size_t)NN * 64;            // N*64

  // 0. init accumulators
  init_att<<<(NN * 256 + 255) / 256, 256, 0, stream>>>(nmax, nsum, oat);

  // 1. node pre-layer: hx = LN(x @ node_W + b)
  node_pre<<<NN / 32, 64, 0, stream>>>(x_s, node_W, node_b, node_lng, node_lnb, hx);

  // 2. q/k/v/skip projections (64 -> 256 each), WMMA + TDM-staged B panel
  dim3 gqkv(NN / 16, 4);
  gemm_bias<<<gqkv, 128, 0, stream>>>(hx, Wq,    bq,    q,  64, 256, 64);
  gemm_bias<<<gqkv, 128, 0, stream>>>(hx, Wk,    bk,    kb, 64, 256, 64);
  gemm_bias<<<gqkv, 128, 0, stream>>>(hx, Wv,    bv,    vb, 64, 256, 64);
  gemm_bias<<<gqkv, 128, 0, stream>>>(hx, Wskip, bskip, xr, 64, 256, 64);

  // 3. fused edge pipeline -> ee
  edge_feat<<<NE / 32, 64, 0, stream>>>(ea, emb, edge_W, edge_b, edge_lng, edge_lnb,
                                        We, be, ee);

  // 4. segment-softmax attention
  att_raw<<<NE * 4 / 256, 256, 0, stream>>>(q, kb, ee, ei, alp, nmax);
  att_exp<<<NE * 4 / 256, 256, 0, stream>>>(ei, nmax, alp, nsum);
  att_scatter<<<NE, 256, 0, stream>>>(vb, ee, alp, nsum, ei, oat);

  // 5. beta gate (writes g into the dead q buffer), then fused post block
  beta_gate<<<NN / 8, 256, 0, stream>>>(oat, xr, Wbeta, q);
  node_post<<<NN / 32, 64, 0, stream>>>(q, lin1_W, lin1_b, ln1_g, ln1_b,
                                        norm_g, norm_b, hx, zin);
  dim3 gz(NN / 16, 1);
  gemm_bias<<<gz, 128, 0, stream>>>(zin, lin_W, lin_b, zz, 64, 64, 64);

  // 6. Set2Set -> d_out (512,128)
  set2set_k<<<NGRAPH, 128, 0, stream>>>(zz, batch, lstm_Wih, lstm_Whh,
                                        lstm_bih, lstm_bhh, (float*)d_out);
}